// CornerActivationB_85220741087981
// MI455X (gfx1250) — compile-verified
//
#include <hip/hip_runtime.h>

typedef __attribute__((ext_vector_type(2))) float v2f;
typedef __attribute__((ext_vector_type(8))) float v8f;

constexpr int GROUPS      = 512;
constexpr int OUT_DIM     = 16;
constexpr int BATCH       = 8192;
constexpr int XCOLS       = GROUPS * 2;        // 1024
constexpr int OCOLS       = GROUPS * OUT_DIM;  // 8192
constexpr int BT_PER_WAVE = 8;                 // 16-row batch tiles per wave
constexpr int WAVES_TOTAL = GROUPS * (BATCH / 16 / BT_PER_WAVE); // 512*64 = 32768
constexpr int WAVES_PER_BLOCK = 8;             // 256 threads, wave32
constexpr int BLOCKS      = WAVES_TOTAL / WAVES_PER_BLOCK;       // 4096

// D = A(16x12,f32) * B(12x16,f32) + C via three V_WMMA_F32_16X16X4_F32 steps.
// A = tent-weight outer products per batch row (K=9 padded to 12 with zeros)
// B = per-group expanded params ep (9x16 padded to 12x16)
__global__ __launch_bounds__(256) void corner_act_wmma(
    const float* __restrict__ X, const float* __restrict__ P, float* __restrict__ O)
{
    const int lane = threadIdx.x & 31;
    const int wave = blockIdx.x * WAVES_PER_BLOCK + (threadIdx.x >> 5);
    const int g    = wave & (GROUPS - 1);   // group (fast dim -> store locality)
    const int sb   = wave >> 9;             // super batch tile [0,64)
    const int n    = lane & 15;             // column / row-in-tile index
    const int h    = lane >> 4;             // wave half selects K pairs

    // ---- per-group expanded params ep[9][16] -> B fragments (once per wave) ----
    // params layout (512,4,16): p[i0][i1] at g*64 + (2*i0+i1)*16 + d
    const float* pg = P + g * 64 + n;
    const float p00 = pg[0], p01 = pg[16], p10 = pg[32], p11 = pg[48];
    // first expansion (dim0): q[a] = interp(a, p[0][*], p[1][*])
    const float q00 = p00,                q01 = p01;
    const float q10 = 0.5f * (p00 + p10), q11 = 0.5f * (p01 + p11);
    const float q20 = p10,                q21 = p11;
    // second expansion (dim1): ep[3a+c] = interp(c, q[a][0], q[a][1])
    float ep[9];
    ep[0] = q00; ep[1] = 0.5f * (q00 + q01); ep[2] = q01;
    ep[3] = q10; ep[4] = 0.5f * (q10 + q11); ep[5] = q11;
    ep[6] = q20; ep[7] = 0.5f * (q20 + q21); ep[8] = q21;
    // B 4x16 f32 layout: lane = N, K = 2*half + vgpr
    const v2f bf0 = { h ? ep[2] : ep[0], h ? ep[3] : ep[1] };
    const v2f bf1 = { h ? ep[6] : ep[4], h ? ep[7] : ep[5] };
    const v2f bf2 = { h ? 0.0f : ep[8], 0.0f };             // K = 8 (9..11 pad)

    const int b_base = sb * (BT_PER_WAVE * 16);
    float* obase = O + (size_t)(b_base + h * 8) * OCOLS + (size_t)g * OUT_DIM + n;

    #pragma unroll
    for (int it = 0; it < BT_PER_WAVE; ++it) {
        const int b0  = b_base + it * 16;
        const int row = b0 + n;                       // this lane's batch row (M)
        const float* xp = X + (size_t)row * XCOLS + 2 * g;
        const float x0 = xp[0];
        const float x1 = xp[1];

        // tent-basis weights over knots {0,1,2} on t = clip(x,-1,1)+1
        const float t0 = fminf(fmaxf(x0, -1.0f), 1.0f) + 1.0f;
        const float t1 = fminf(fmaxf(x1, -1.0f), 1.0f) + 1.0f;
        const float w00 = fmaxf(0.0f, 1.0f - fabsf(t0));
        const float w01 = fmaxf(0.0f, 1.0f - fabsf(t0 - 1.0f));
        const float w02 = fmaxf(0.0f, 1.0f - fabsf(t0 - 2.0f));
        const float w10 = fmaxf(0.0f, 1.0f - fabsf(t1));
        const float w11 = fmaxf(0.0f, 1.0f - fabsf(t1 - 1.0f));
        const float w12 = fmaxf(0.0f, 1.0f - fabsf(t1 - 2.0f));
        // wk[3*j0 + j1] = w0[j0] * w1[j1]
        const float wk0 = w00 * w10, wk1 = w00 * w11, wk2 = w00 * w12;
        const float wk3 = w01 * w10, wk4 = w01 * w11, wk5 = w01 * w12;
        const float wk6 = w02 * w10, wk7 = w02 * w11, wk8 = w02 * w12;
        // A 16x4 f32 layout: lane&15 = M, K = 2*half + vgpr
        const v2f af0 = { h ? wk2 : wk0, h ? wk3 : wk1 };
        const v2f af1 = { h ? wk6 : wk4, h ? wk7 : wk5 };
        const v2f af2 = { h ? 0.0f : wk8, 0.0f };

        v8f c = {};
        c = __builtin_amdgcn_wmma_f32_16x16x4_f32(false, af0, false, bf0,
                                                  (short)0, c, false, false);
        c = __builtin_amdgcn_wmma_f32_16x16x4_f32(false, af1, false, bf1,
                                                  (short)0, c, false, false);
        c = __builtin_amdgcn_wmma_f32_16x16x4_f32(false, af2, false, bf2,
                                                  (short)0, c, false, false);

        // D 16x16 f32: VGPR r -> row b0 + r + 8h, col = g*16 + n
        float* orow = obase + (size_t)(it * 16) * OCOLS;
        #pragma unroll
        for (int r = 0; r < 8; ++r)
            orow[(size_t)r * OCOLS] = c[r];
    }
}

extern "C" void kernel_launch(void* const* d_in, const int* in_sizes, int n_in,
                              void* d_out, int out_size, void* d_ws, size_t ws_size,
                              hipStream_t stream)
{
    const float* X = (const float*)d_in[0];   // (8192, 1024) f32
    const float* P = (const float*)d_in[1];   // (512, 4, 16) f32
    float* O       = (float*)d_out;           // (8192, 8192) f32

    corner_act_wmma<<<BLOCKS, 256, 0, stream>>>(X, P, O);
}